// RelPosSelfAttention_31499290148950
// MI455X (gfx1250) — compile-verified
//
#include <hip/hip_runtime.h>
#include <hip/hip_bf16.h>

// ---------------------------------------------------------------------------
// RelPos self-attention (Transformer-XL style), MI455X / gfx1250, wave32.
//
// B=64, M=2048, C=1024, H=16, D=64, L=M+1=2049.
// Fold Wk/Wv through the attention (saves ~550 GFLOP of K/V projection);
// the two big passes stream `mem` (512 MB) once each => HBM-bound
// (~44us at 23.3 TB/s). Both passes are 16-wide GEMMs on
// V_WMMA_F32_16X16X4_F32, with cat tiles DMA'd into LDS by the Tensor Data
// Mover (double-buffered, TENSORcnt-synchronized). The l==M row (from x) and
// l>M padding are peeled out of the streaming kernels so every TDM tile is a
// full interior 64x64 tile of `mem`.
// ---------------------------------------------------------------------------

typedef float    v2f  __attribute__((ext_vector_type(2)));
typedef float    v8f  __attribute__((ext_vector_type(8)));
typedef unsigned u32x4 __attribute__((ext_vector_type(4)));
typedef int      i32x4 __attribute__((ext_vector_type(4)));
typedef int      i32x8 __attribute__((ext_vector_type(8)));

constexpr int   B_  = 64;
constexpr int   M_  = 2048;
constexpr int   C_  = 1024;
constexpr int   H_  = 16;
constexpr int   D_  = 64;
constexpr int   L_  = M_ + 1;       // 2049
constexpr int   LT_ = 2112;         // L padded to 33*64
constexpr float SCALE_ = 0.125f;    // 1/sqrt(D)

// ---- workspace layout (floats), each region fully rewritten every call ----
constexpr size_t alignup(size_t v) { return (v + 255) & ~size_t(255); }
constexpr size_t OFF_R   = 0;                                          // L*D
constexpr size_t OFF_Q   = alignup(OFF_R   + (size_t)L_ * D_);         // B*C
constexpr size_t OFF_QKT = alignup(OFF_Q   + (size_t)B_ * C_);         // B*C*H
constexpr size_t OFF_S   = alignup(OFF_QKT + (size_t)B_ * C_ * H_);    // B*H*LT
constexpr size_t OFF_CTX = alignup(OFF_S   + (size_t)B_ * H_ * LT_);   // B*H*C
constexpr size_t OFF_HO  = alignup(OFF_CTX + (size_t)B_ * H_ * C_);    // B*C

// ---------------------------------------------------------------------------
// TDM: DMA a 64x64 f32 tile (row stride `stride0` floats) from global into
// LDS at byte offset `lds_off`, with 1-dword padding after every 64 dwords
// so the LDS image has stride 65 floats (bank-conflict-free fragment reads).
// D# bitfields per CDNA5 ISA. Issued by wave 0 only.
// This toolchain's builtin is the 6-arg clang-23 form:
//   (u32x4 g0, i32x8 g1, i32x4 g2, i32x4 g3, i32x8 unused, i32 cpol)
// ---------------------------------------------------------------------------
__device__ __forceinline__ void tdm_load_tile64(unsigned lds_off,
                                                const float* gaddr,
                                                unsigned tensor_d0,
                                                unsigned tensor_d1,
                                                unsigned stride0) {
  unsigned long long ga = (unsigned long long)gaddr;
  u32x4 g0;
  g0[0] = 1u;                                   // count=1, user descriptor
  g0[1] = lds_off;                              // lds_addr (bytes)
  g0[2] = (unsigned)(ga & 0xFFFFFFFFu);         // global_addr[31:0]
  g0[3] = (unsigned)((ga >> 32) & 0x1FFFFFFu)   // global_addr[56:32]
        | (2u << 30);                           // type=2 ("image")
  unsigned w0 = (2u << 16)                      // data_size: 4 bytes
              | (1u << 20)                      // pad_enable
              | (5u << 22)                      // pad_interval: 64 dwords
              | (0u << 25);                     // pad_amount: 1 dword
  unsigned w1 = (tensor_d0 & 0xFFFFu) << 16;                  // dim0[15:0] @48
  unsigned w2 = (tensor_d0 >> 16) | ((tensor_d1 & 0xFFFFu) << 16);
  unsigned w3 = (tensor_d1 >> 16) | (64u << 16);              // tile_dim0=64
  unsigned w4 = 64u;                                          // tile_dim1=64
  unsigned w5 = stride0;                                      // dim0_stride lo
  i32x8 g1 = {(int)w0, (int)w1, (int)w2, (int)w3, (int)w4, (int)w5, 0, 0};
  i32x4 g2 = {1, 1, 0, 0};                      // dims 2/3 unused (tile_dim2=0)
  i32x4 g3 = {0, 0, 0, 0};
  i32x8 g4 = {0, 0, 0, 0, 0, 0, 0, 0};          // unused trailing group
  __builtin_amdgcn_tensor_load_to_lds(g0, g1, g2, g3, g4, 0);
}

// ---------------------------------------------------------------------------
// 0) sinusoidal table r[l][d] = emb[L-1-l][d]
// ---------------------------------------------------------------------------
__global__ void k_rtable(float* __restrict__ r) {
  int idx = blockIdx.x * 256 + threadIdx.x;
  if (idx >= L_ * D_) return;
  int l = idx / D_, d = idx % D_;
  float p    = (float)(L_ - 1 - l);
  float freq = __powf(10000.0f, -(float)(d & 31) / 32.0f);
  float ang  = p * freq;
  r[idx] = (d < 32) ? __sinf(ang) : __cosf(ang);
}

// ---------------------------------------------------------------------------
// 1) q = x @ Wq + bq
// ---------------------------------------------------------------------------
__global__ void k_qproj(const float* __restrict__ x, const float* __restrict__ Wq,
                        const float* __restrict__ bq, float* __restrict__ q) {
  int idx = blockIdx.x * 256 + threadIdx.x;
  int b = idx >> 10, c = idx & 1023;
  const float* xr = x + (size_t)b * C_;
  float acc = bq[c];
  for (int k = 0; k < C_; ++k) acc = fmaf(xr[k], Wq[(size_t)k * C_ + c], acc);
  q[idx] = acc;
}

// ---------------------------------------------------------------------------
// 2) fold Wk into query: qkT[b][cin][h] = sum_d (q+u)[b,h,d] * Wk[cin, h*D+d]
// ---------------------------------------------------------------------------
__global__ void k_qkfold(const float* __restrict__ q, const float* __restrict__ u,
                         const float* __restrict__ Wk, float* __restrict__ qkT) {
  int idx = blockIdx.x * 256 + threadIdx.x;
  int h = idx & 15, cin = (idx >> 4) & 1023, b = idx >> 14;
  const float* qr = q + (size_t)b * C_ + h * D_;
  const float* ur = u + h * D_;
  const float* wr = Wk + (size_t)cin * C_ + h * D_;
  float acc = 0.f;
  #pragma unroll 4
  for (int d = 0; d < D_; ++d) acc = fmaf(qr[d] + ur[d], wr[d], acc);
  qkT[idx] = acc;
}

// ---------------------------------------------------------------------------
// 3) bd term (pre-scaled) into score buffer; -1e30 mask for l >= L
// ---------------------------------------------------------------------------
__global__ void k_bd(const float* __restrict__ q, const float* __restrict__ vb,
                     const float* __restrict__ r, float* __restrict__ s) {
  int b = blockIdx.x >> 4, h = blockIdx.x & 15;
  __shared__ float qv[D_];
  if (threadIdx.x < D_)
    qv[threadIdx.x] = q[(size_t)b * C_ + h * D_ + threadIdx.x] + vb[h * D_ + threadIdx.x];
  __syncthreads();
  float* srow = s + (size_t)(b * H_ + h) * LT_;
  for (int l = threadIdx.x; l < LT_; l += 256) {
    if (l >= L_) { srow[l] = -1e30f; continue; }
    const float* rr = r + (size_t)l * D_;
    float acc = 0.f;
    #pragma unroll 4
    for (int d = 0; d < D_; ++d) acc = fmaf(qv[d], rr[d], acc);
    srow[l] = acc * SCALE_;
  }
}

// ---------------------------------------------------------------------------
// 3b) peeled l==M score row: s[b,h,M] += scale * (x[b]·qkT[b,:,h])
// ---------------------------------------------------------------------------
__global__ void k_xscore(const float* __restrict__ x, const float* __restrict__ qkT,
                         float* __restrict__ s) {
  int idx = blockIdx.x * 256 + threadIdx.x;     // B*H threads
  int b = idx >> 4, h = idx & 15;
  const float* xr = x + (size_t)b * C_;
  const float* qc = qkT + (size_t)b * C_ * H_ + h;
  float acc = 0.f;
  for (int c = 0; c < C_; ++c) acc = fmaf(xr[c], qc[(size_t)c * H_], acc);
  size_t off = (size_t)(b * H_ + h) * LT_ + M_;
  s[off] = acc * SCALE_ + s[off];
}

// ---------------------------------------------------------------------------
// 4) scores pass (mem read #1): S[b, l(16), h(16)] via WMMA f32 16x16x4.
//    TDM double-buffers 64x64 cat tiles into LDS; qkT chunk (4KB) staged by
//    the threads into the other LDS buffer of the same pipeline stage.
//    grid (32, 64) -- all tiles interior to mem. 128 threads = 4 waves.
// ---------------------------------------------------------------------------
__global__ __launch_bounds__(128) void k_scores(const float* __restrict__ mem,
                                                const float* __restrict__ qkT,
                                                float* __restrict__ s) {
  __shared__ float cs[2][64][65];
  __shared__ float qs[2][1024];                 // [k(64)][h(16)]
  const int b    = blockIdx.y;
  const int l0   = blockIdx.x * 64;
  const int wave = threadIdx.x >> 5;
  const int lane = threadIdx.x & 31;
  const int m    = lane & 15;                   // A row (l) == B col (h)
  const int kb   = (lane >> 4) << 1;            // K sub-offset {0,2}
  const float* gbase = mem + ((size_t)b * M_ + l0) * C_;
  const float4* qbase = (const float4*)(qkT + (size_t)b * C_ * H_);
  v8f acc = {0.f, 0.f, 0.f, 0.f, 0.f, 0.f, 0.f, 0.f};

  // ---- stage chunk 0 ----
  if (threadIdx.x < 32)
    tdm_load_tile64((unsigned)(unsigned long long)&cs[0][0][0], gbase, C_, M_, C_);
  {
    #pragma unroll
    for (int i = 0; i < 2; ++i) {               // qkT chunk 0: 1024 floats
      int idx = threadIdx.x + i * 128;
      float4 v = qbase[idx];
      ((float4*)qs[0])[idx] = v;
    }
  }
  __builtin_amdgcn_s_wait_tensorcnt(0);
  __syncthreads();

  for (int kc = 0; kc < 16; ++kc) {
    const int cur = kc & 1;
    if (kc + 1 < 16) {                          // prefetch next chunk
      if (threadIdx.x < 32)
        tdm_load_tile64((unsigned)(unsigned long long)&cs[cur ^ 1][0][0],
                        gbase + (kc + 1) * 64, C_, M_, C_);
      #pragma unroll
      for (int i = 0; i < 2; ++i) {
        int idx = threadIdx.x + i * 128;
        float4 v = qbase[(kc + 1) * 256 + idx];
        ((float4*)qs[cur ^ 1])[idx] = v;
      }
    }
    #pragma unroll
    for (int kk = 0; kk < 16; ++kk) {
      int k = kk * 4 + kb;
      v2f a;  a.x  = cs[cur][wave * 16 + m][k];  a.y  = cs[cur][wave * 16 + m][k + 1];
      v2f bm; bm.x = qs[cur][k * 16 + m];        bm.y = qs[cur][(k + 1) * 16 + m];
      acc = __builtin_amdgcn_wmma_f32_16x16x4_f32(false, a, false, bm,
                                                  (short)0, acc, false, false);
    }
    __builtin_amdgcn_s_wait_tensorcnt(0);       // no-op for non-issuing waves
    __syncthreads();
  }

  const int mrow = (lane >> 4) * 8;             // D rows j | j+8 per lane half
  #pragma unroll
  for (int j = 0; j < 8; ++j) {
    int l = l0 + wave * 16 + mrow + j;
    size_t off = (size_t)(b * H_ + m) * LT_ + l;
    s[off] = acc[j] * SCALE_ + s[off];          // += pre-scaled bd term
  }
}

// ---------------------------------------------------------------------------
// 5) softmax over l per (b,h); masked entries -> exactly 0
// ---------------------------------------------------------------------------
__global__ void k_softmax(float* __restrict__ s) {
  int b = blockIdx.x >> 4, h = blockIdx.x & 15;
  float* row = s + (size_t)(b * H_ + h) * LT_;
  __shared__ float red[256];
  float mx = -3.4e38f;
  for (int l = threadIdx.x; l < LT_; l += 256) mx = fmaxf(mx, row[l]);
  red[threadIdx.x] = mx; __syncthreads();
  for (int st = 128; st > 0; st >>= 1) {
    if (threadIdx.x < st) red[threadIdx.x] = fmaxf(red[threadIdx.x], red[threadIdx.x + st]);
    __syncthreads();
  }
  mx = red[0]; __syncthreads();
  float sum = 0.f;
  for (int l = threadIdx.x; l < LT_; l += 256) {
    float e = __expf(row[l] - mx);
    row[l] = e; sum += e;
  }
  red[threadIdx.x] = sum; __syncthreads();
  for (int st = 128; st > 0; st >>= 1) {
    if (threadIdx.x < st) red[threadIdx.x] += red[threadIdx.x + st];
    __syncthreads();
  }
  float inv = 1.f / red[0];
  for (int l = threadIdx.x; l < LT_; l += 256) row[l] *= inv;
}

// ---------------------------------------------------------------------------
// 6) ctx pass (mem read #2): ctx[b, h(16), c(16)] = p @ cat over l.
//    TDM double-buffers cat tiles; p chunks staged by threads. The l==M term
//    (p[b,h,M] * x[b,c]) is added at store time, so K covers only mem rows.
// ---------------------------------------------------------------------------
__global__ __launch_bounds__(128) void k_ctx(const float* __restrict__ mem,
                                             const float* __restrict__ x,
                                             const float* __restrict__ p,
                                             float* __restrict__ ctx) {
  __shared__ float cs[2][64][65];
  __shared__ float ps[2][16][65];
  const int b    = blockIdx.y;
  const int c0   = blockIdx.x * 64;
  const int wave = threadIdx.x >> 5;
  const int lane = threadIdx.x & 31;
  const int n    = lane & 15;                   // A row (h) == B col (c)
  const int kb   = (lane >> 4) << 1;
  const float* gbase = mem + (size_t)b * M_ * C_ + c0;
  v8f acc = {0.f, 0.f, 0.f, 0.f, 0.f, 0.f, 0.f, 0.f};

  // ---- stage chunk 0 ----
  if (threadIdx.x < 32)
    tdm_load_tile64((unsigned)(unsigned long long)&cs[0][0][0], gbase, C_, M_, C_);
  {
    #pragma unroll
    for (int i = 0; i < 2; ++i) {               // p chunk 0: 16 heads x 64 l
      int idx = threadIdx.x + i * 128;
      int h = idx >> 4, col = (idx & 15) << 2;
      float4 v = *(const float4*)(p + (size_t)(b * H_ + h) * LT_ + col);
      ps[0][h][col + 0] = v.x; ps[0][h][col + 1] = v.y;
      ps[0][h][col + 2] = v.z; ps[0][h][col + 3] = v.w;
    }
  }
  __builtin_amdgcn_s_wait_tensorcnt(0);
  __syncthreads();

  for (int kc = 0; kc < 32; ++kc) {             // 32 interior chunks of 64 l
    const int cur = kc & 1;
    if (kc + 1 < 32) {
      if (threadIdx.x < 32)
        tdm_load_tile64((unsigned)(unsigned long long)&cs[cur ^ 1][0][0],
                        gbase + (size_t)(kc + 1) * 64 * C_, C_, M_, C_);
      #pragma unroll
      for (int i = 0; i < 2; ++i) {
        int idx = threadIdx.x + i * 128;
        int h = idx >> 4, col = (idx & 15) << 2;
        float4 v = *(const float4*)(p + (size_t)(b * H_ + h) * LT_ + (kc + 1) * 64 + col);
        ps[cur ^ 1][h][col + 0] = v.x; ps[cur ^ 1][h][col + 1] = v.y;
        ps[cur ^ 1][h][col + 2] = v.z; ps[cur ^ 1][h][col + 3] = v.w;
      }
    }
    #pragma unroll
    for (int kk = 0; kk < 16; ++kk) {
      int k = kk * 4 + kb;
      v2f a;  a.x  = ps[cur][n][k];              a.y  = ps[cur][n][k + 1];
      v2f bm; bm.x = cs[cur][k][wave * 16 + n];  bm.y = cs[cur][k + 1][wave * 16 + n];
      acc = __builtin_amdgcn_wmma_f32_16x16x4_f32(false, a, false, bm,
                                                  (short)0, acc, false, false);
    }
    __builtin_amdgcn_s_wait_tensorcnt(0);
    __syncthreads();
  }

  const int mrow = (lane >> 4) * 8;
  const float xv = x[(size_t)b * C_ + c0 + wave * 16 + n];
  #pragma unroll
  for (int j = 0; j < 8; ++j) {
    int h = mrow + j;
    float pt = p[(size_t)(b * H_ + h) * LT_ + M_];   // l==M weight
    ctx[(size_t)(b * H_ + h) * C_ + c0 + wave * 16 + n] = acc[j] + pt * xv;
  }
}

// ---------------------------------------------------------------------------
// 7) ho[b, h*D+d] = ctx[b,h,:] @ Wv[:, h*D+d] + bv
// ---------------------------------------------------------------------------
__global__ void k_vproj(const float* __restrict__ ctx, const float* __restrict__ Wv,
                        const float* __restrict__ bv, float* __restrict__ ho) {
  int idx = blockIdx.x * 256 + threadIdx.x;
  int b = idx >> 10, co = idx & 1023, h = co >> 6;
  const float* cr = ctx + (size_t)(b * H_ + h) * C_;
  float acc = bv[co];
  for (int k = 0; k < C_; ++k) acc = fmaf(cr[k], Wv[(size_t)k * C_ + co], acc);
  ho[idx] = acc;
}

// ---------------------------------------------------------------------------
// 8) out = ho @ Wo + bo
// ---------------------------------------------------------------------------
__global__ void k_oproj(const float* __restrict__ ho, const float* __restrict__ Wo,
                        const float* __restrict__ bo, float* __restrict__ out) {
  int idx = blockIdx.x * 256 + threadIdx.x;
  int b = idx >> 10, co = idx & 1023;
  const float* hr = ho + (size_t)b * C_;
  float acc = bo[co];
  for (int k = 0; k < C_; ++k) acc = fmaf(hr[k], Wo[(size_t)k * C_ + co], acc);
  out[idx] = acc;
}

// ---------------------------------------------------------------------------
extern "C" void kernel_launch(void* const* d_in, const int* in_sizes, int n_in,
                              void* d_out, int out_size, void* d_ws, size_t ws_size,
                              hipStream_t stream) {
  const float* x   = (const float*)d_in[0];
  const float* mem = (const float*)d_in[1];
  const float* Wq  = (const float*)d_in[2];
  const float* bq  = (const float*)d_in[3];
  const float* Wk  = (const float*)d_in[4];
  // d_in[5] = bk: constant per (b,h) across l -> cancels in softmax, unused
  const float* Wv  = (const float*)d_in[6];
  const float* bv  = (const float*)d_in[7];
  const float* ub  = (const float*)d_in[8];
  const float* vb  = (const float*)d_in[9];
  const float* Wo  = (const float*)d_in[10];
  const float* bo  = (const float*)d_in[11];
  float* out = (float*)d_out;

  float* ws  = (float*)d_ws;
  float* r   = ws + OFF_R;
  float* q   = ws + OFF_Q;
  float* qkT = ws + OFF_QKT;
  float* s   = ws + OFF_S;
  float* ctx = ws + OFF_CTX;
  float* ho  = ws + OFF_HO;

  k_rtable <<<(L_ * D_ + 255) / 256, 256, 0, stream>>>(r);
  k_qproj  <<<(B_ * C_) / 256,       256, 0, stream>>>(x, Wq, bq, q);
  k_qkfold <<<(B_ * C_ * H_) / 256,  256, 0, stream>>>(q, ub, Wk, qkT);
  k_bd     <<<B_ * H_,               256, 0, stream>>>(q, vb, r, s);
  k_xscore <<<(B_ * H_) / 256,       256, 0, stream>>>(x, qkT, s);
  k_scores <<<dim3(M_ / 64, B_),     128, 0, stream>>>(mem, qkT, s);
  k_softmax<<<B_ * H_,               256, 0, stream>>>(s);
  k_ctx    <<<dim3(C_ / 64, B_),     128, 0, stream>>>(mem, x, s, ctx);
  k_vproj  <<<(B_ * C_) / 256,       256, 0, stream>>>(ctx, Wv, bv, ho);
  k_oproj  <<<(B_ * C_) / 256,       256, 0, stream>>>(ho, Wo, bo, out);
}